// Generator_54013508714861
// MI455X (gfx1250) — compile-verified
//
#include <hip/hip_runtime.h>
#include <math.h>

// Problem constants (from the reference).
#define NSZ 90     // source/feature dim, also node count
#define NTZ 120    // target feature dim
#define EZ  (NSZ * NSZ)
#define P96 96     // padded 90
#define P128 128   // padded 120
#define NTHREADS 256
#define NWAVES 8   // wave32: 256 threads = 8 waves

typedef float v2f __attribute__((ext_vector_type(2)));
typedef float v8f __attribute__((ext_vector_type(8)));

// ---- fp32 WMMA fragment loaders (layouts per CDNA5 ISA §7.12.2) ----
// A (16x4): lane holds row M=tm+(lane&15); v0/v1 = K = k0+2*(lane>>4) + {0,1}
__device__ __forceinline__ v2f lds_a_frag(const float* S, int ld, int tm, int k0, int lane) {
  const int h = lane >> 4;
  const int r = tm + (lane & 15);
  const int k = k0 + 2 * h;
  v2f a;
  a.x = S[r * ld + k];
  a.y = S[r * ld + k + 1];
  return a;
}
// B (4x16): lane holds col N=tn+(lane&15); v0/v1 = rows K = k0+2*(lane>>4) + {0,1}
__device__ __forceinline__ v2f lds_b_frag(const float* S, int ld, int tn, int k0, int lane) {
  const int h = lane >> 4;
  const int n = tn + (lane & 15);
  const int k = k0 + 2 * h;
  v2f b;
  b.x = S[k * ld + n];
  b.y = S[(k + 1) * ld + n];
  return b;
}
// B fragment straight from global row-major [rows x cols], zero-padded, optional relu.
__device__ __forceinline__ v2f glb_b_frag(const float* __restrict__ W, int rows, int cols,
                                          int tn, int k0, int lane, bool relu) {
  const int h = lane >> 4;
  const int n = tn + (lane & 15);
  const int k = k0 + 2 * h;
  float b0 = 0.f, b1 = 0.f;
  if (n < cols) {
    if (k < rows)     b0 = W[k * cols + n];
    if (k + 1 < rows) b1 = W[(k + 1) * cols + n];
  }
  if (relu) {
    b0 = b0 > 0.f ? b0 : 0.f;
    b1 = b1 > 0.f ? b1 : 0.f;
  }
  v2f b;
  b.x = b0;
  b.y = b1;
  return b;
}

// Generic tiled GEMM over 16x16 output tiles; tiles distributed across waves.
// Control flow is wave-uniform so EXEC is all-ones at every WMMA (ISA §7.12 requirement).
template <typename LA, typename LB, typename SC>
__device__ __forceinline__ void wmma_gemm(int mtiles, int ntiles, int ksteps,
                                          int wave, LA la, LB lb, SC sc) {
  const int total = mtiles * ntiles;
  for (int t = wave; t < total; t += NWAVES) {
    const int tm = (t / ntiles) * 16;
    const int tn = (t % ntiles) * 16;
    v8f c = {0.f, 0.f, 0.f, 0.f, 0.f, 0.f, 0.f, 0.f};
    for (int ks = 0; ks < ksteps; ++ks) {
      v2f a = la(tm, ks * 4);
      v2f b = lb(tn, ks * 4);
      c = __builtin_amdgcn_wmma_f32_16x16x4_f32(false, a, false, b, (short)0, c,
                                                false, false);
    }
    sc(tm, tn, c);
  }
}

__global__ void __launch_bounds__(NTHREADS)
brain_gnn_fused(const float* __restrict__ x,
                const float* __restrict__ ea,
                const float* __restrict__ Wm1,
                const float* __restrict__ root1,
                const float* __restrict__ bias1,
                const float* __restrict__ gamma1,
                const float* __restrict__ beta1,
                const float* __restrict__ Wm3,
                const float* __restrict__ root3,
                const float* __restrict__ bias3,
                const float* __restrict__ gamma3,
                const float* __restrict__ beta3,
                float* __restrict__ out) {
  // 156 KB static LDS (WGP has 320 KB).
  __shared__ float sX[P96 * P96];    // x, later x1 (padded, pads stay 0)
  __shared__ float sAT[P96 * P96];   // A^T: AT[dst][src] = edge_attr[src*90+dst]
  __shared__ float sY[P96 * P96];    // (A^T x) / 90
  __shared__ float sH[P96 * P128];   // h1 then h3/x3

  const int tid = threadIdx.x;
  const int lane = tid & 31;
  const int wave = tid >> 5;

  // Warm L2 for the weight matrices we stream as B-fragments (global_prefetch_b8).
  if (wave == 0) {
    __builtin_prefetch(Wm1 + lane * 256, 0, 1);
    __builtin_prefetch(root1 + lane * 256, 0, 1);
    __builtin_prefetch(Wm3 + lane * 336, 0, 1);
    __builtin_prefetch(root3 + lane * 336, 0, 1);
  }

  // Zero the padded buffers whose pads must be exactly 0.
  for (int i = tid; i < P96 * P96; i += NTHREADS) { sX[i] = 0.f; sAT[i] = 0.f; }
  for (int i = tid; i < P96 * P128; i += NTHREADS) sH[i] = 0.f;
  __syncthreads();

  // Stage x and A^T into LDS.
  for (int e = tid; e < EZ; e += NTHREADS) {
    const int r = e / NSZ;          // src node i
    const int c = e - r * NSZ;      // dst node j / feature
    sX[r * P96 + c] = x[e];
    sAT[c * P96 + r] = ea[e];       // AT[j][i] = a[i*90+j]
  }
  __syncthreads();

  // ---- Stage 1: sY = (A^T @ X) * (1/90)   [96x96, K=96] ----
  wmma_gemm(6, 6, 24, wave,
    [&](int tm, int k0) { return lds_a_frag(sAT, P96, tm, k0, lane); },
    [&](int tn, int k0) { return lds_b_frag(sX, P96, tn, k0, lane); },
    [&](int tm, int tn, v8f c) {
      const int h = lane >> 4, n = tn + (lane & 15);
      const float inv = 1.0f / 90.0f;
      for (int r = 0; r < 8; ++r) sY[(tm + r + 8 * h) * P96 + n] = c[r] * inv;
    });
  __syncthreads();

  // ---- Stage 2: H1 = sY @ relu(W1) + X @ root1 + bias1   [rows<90] ----
  wmma_gemm(6, 6, 48, wave,
    [&](int tm, int k0) {
      return (k0 < P96) ? lds_a_frag(sY, P96, tm, k0, lane)
                        : lds_a_frag(sX, P96, tm, k0 - P96, lane);
    },
    [&](int tn, int k0) {
      return (k0 < P96) ? glb_b_frag(Wm1, NSZ, NSZ, tn, k0, lane, true)
                        : glb_b_frag(root1, NSZ, NSZ, tn, k0 - P96, lane, false);
    },
    [&](int tm, int tn, v8f c) {
      const int h = lane >> 4, n = tn + (lane & 15);
      const float bv = (n < NSZ) ? bias1[n] : 0.f;
      for (int r = 0; r < 8; ++r) {
        const int m = tm + r + 8 * h;
        if (m < NSZ) sH[m * P128 + n] = c[r] + bv;
      }
    });
  __syncthreads();

  // ---- BatchNorm1 (biased var, eps=1e-3) + sigmoid -> sX (x1) ----
  if (tid < NSZ) {
    const int c = tid;
    float s = 0.f;
    for (int r = 0; r < NSZ; ++r) s += sH[r * P128 + c];
    const float mu = s * (1.0f / 90.0f);
    float v = 0.f;
    for (int r = 0; r < NSZ; ++r) { const float d = sH[r * P128 + c] - mu; v += d * d; }
    v *= (1.0f / 90.0f);
    const float sc = gamma1[c] * rsqrtf(v + 1e-3f);
    const float sh = beta1[c] - mu * sc;
    for (int r = 0; r < NSZ; ++r) {
      const float z = sH[r * P128 + c] * sc + sh;
      sX[r * P96 + c] = 1.0f / (1.0f + expf(-z));
    }
  }
  __syncthreads();

  // ---- Stage 3: sY = (A^T @ X1) * (1/90) ----
  wmma_gemm(6, 6, 24, wave,
    [&](int tm, int k0) { return lds_a_frag(sAT, P96, tm, k0, lane); },
    [&](int tn, int k0) { return lds_b_frag(sX, P96, tn, k0, lane); },
    [&](int tm, int tn, v8f c) {
      const int h = lane >> 4, n = tn + (lane & 15);
      const float inv = 1.0f / 90.0f;
      for (int r = 0; r < 8; ++r) sY[(tm + r + 8 * h) * P96 + n] = c[r] * inv;
    });
  __syncthreads();

  // ---- Stage 4: H3 = sY @ relu(W3) + X1 @ root3 + bias3   [90x120 region] ----
  wmma_gemm(6, 8, 48, wave,
    [&](int tm, int k0) {
      return (k0 < P96) ? lds_a_frag(sY, P96, tm, k0, lane)
                        : lds_a_frag(sX, P96, tm, k0 - P96, lane);
    },
    [&](int tn, int k0) {
      return (k0 < P96) ? glb_b_frag(Wm3, NSZ, NTZ, tn, k0, lane, true)
                        : glb_b_frag(root3, NSZ, NTZ, tn, k0 - P96, lane, false);
    },
    [&](int tm, int tn, v8f c) {
      const int h = lane >> 4, n = tn + (lane & 15);
      const float bv = (n < NTZ) ? bias3[n] : 0.f;
      for (int r = 0; r < 8; ++r) {
        const int m = tm + r + 8 * h;
        if (m < NSZ) sH[m * P128 + n] = c[r] + bv;
      }
    });
  __syncthreads();

  // ---- BatchNorm3 + sigmoid, in place in sH (x3) ----
  if (tid < NTZ) {
    const int c = tid;
    float s = 0.f;
    for (int r = 0; r < NSZ; ++r) s += sH[r * P128 + c];
    const float mu = s * (1.0f / 90.0f);
    float v = 0.f;
    for (int r = 0; r < NSZ; ++r) { const float d = sH[r * P128 + c] - mu; v += d * d; }
    v *= (1.0f / 90.0f);
    const float sc = gamma3[c] * rsqrtf(v + 1e-3f);
    const float sh = beta3[c] - mu * sc;
    for (int r = 0; r < NSZ; ++r) {
      const float z = sH[r * P128 + c] * sc + sh;
      sH[r * P128 + c] = 1.0f / (1.0f + expf(-z));
    }
  }
  __syncthreads();

  // ---- Stage 5: out = x3^T @ x3   [120x120, K over 96 node rows] ----
  wmma_gemm(8, 8, 24, wave,
    [&](int tm, int k0) {  // A[m][k] = x3[k][m] (transposed read from LDS)
      const int h = lane >> 4;
      const int m = tm + (lane & 15);
      const int k = k0 + 2 * h;
      v2f a;
      a.x = sH[k * P128 + m];
      a.y = sH[(k + 1) * P128 + m];
      return a;
    },
    [&](int tn, int k0) { return lds_b_frag(sH, P128, tn, k0, lane); },
    [&](int tm, int tn, v8f c) {
      const int h = lane >> 4, n = tn + (lane & 15);
      if (n < NTZ) {
        for (int r = 0; r < 8; ++r) {
          const int m = tm + r + 8 * h;
          if (m < NTZ) out[m * NTZ + n] = c[r];
        }
      }
    });
}

extern "C" void kernel_launch(void* const* d_in, const int* in_sizes, int n_in,
                              void* d_out, int out_size, void* d_ws, size_t ws_size,
                              hipStream_t stream) {
  // setup_inputs() order:
  // 0:x 1:edge_index 2:edge_attr 3:W_mlp1 4:b_mlp1(=0) 5:root1 6:bias1 7:gamma1
  // 8:beta1 9:W_mlp3 10:b_mlp3(=0) 11:root3 12:bias3 13:gamma3 14:beta3
  const float* x     = (const float*)d_in[0];
  const float* ea    = (const float*)d_in[2];
  const float* Wm1   = (const float*)d_in[3];
  const float* root1 = (const float*)d_in[5];
  const float* bias1 = (const float*)d_in[6];
  const float* g1    = (const float*)d_in[7];
  const float* b1    = (const float*)d_in[8];
  const float* Wm3   = (const float*)d_in[9];
  const float* root3 = (const float*)d_in[11];
  const float* bias3 = (const float*)d_in[12];
  const float* g3    = (const float*)d_in[13];
  const float* b3    = (const float*)d_in[14];
  float* out = (float*)d_out;

  brain_gnn_fused<<<dim3(1), dim3(NTHREADS), 0, stream>>>(
      x, ea, Wm1, root1, bias1, g1, b1, Wm3, root3, bias3, g3, b3, out);

  (void)in_sizes; (void)n_in; (void)out_size; (void)d_ws; (void)ws_size;
}